// YoloV3Layer_1984274891274
// MI455X (gfx1250) — compile-verified
//
#include <hip/hip_runtime.h>

// YOLOv3 head: [B, 255, 76, 76] -> [B, 76*76*3, 85] with sigmoid/exp transforms.
// Memory-bound single-pass stream (~189 MB @ 23.3 TB/s ~ 8 us floor).
// Strategy: LDS tile transpose (coalesced b128 NT reads along g, coalesced NT
// writes along c), conflict-free odd-stride LDS layout, fast v_exp/v_rcp math.

#define GDIM       76
#define G2         (GDIM * GDIM)   // 5776
#define NATTR      85
#define NANCH      3
#define NCHAN      (NATTR * NANCH) // 255
#define TBG        32              // grid cells per block tile
#define LDS_STRIDE 33              // odd stride -> bank-conflict-free transpose
#define BLOCK      256
#define STRIDE_F   8.0f            // 608 / 76
#define LOG2E      1.44269504088896340736f

typedef float v4f __attribute__((ext_vector_type(4)));  // native vector for NT builtins

__global__ __launch_bounds__(BLOCK) void yolo_head_kernel(
    const float* __restrict__ x, float* __restrict__ out, int batches) {
  __shared__ float tile[NCHAN * LDS_STRIDE];  // 255*33*4 = 33,660 B

  const int b      = blockIdx.y;
  const int g_base = blockIdx.x * TBG;
  const int t      = threadIdx.x;
  if (b >= batches) return;

  // ---------------- load phase: coalesced float4 NT reads along g ------------
  // 32 rows (channels) per pass x 8 float4 quads covering 32 g's.
  {
    const int r0 = t >> 3;              // row (channel) within pass: 0..31
    const int q  = t & 7;               // float4 quad: 0..7
    const int gq = g_base + q * 4;      // g offset of this quad
    if (gq < G2) {                      // G2 % 4 == 0 -> quad fully valid
      const size_t bbase = (size_t)b * NCHAN * G2;
      #pragma unroll
      for (int c = r0; c < NCHAN; c += 32) {
        const v4f* src =
            reinterpret_cast<const v4f*>(x + bbase + (size_t)c * G2 + gq);
        v4f v = __builtin_nontemporal_load(src);
        float* dst = &tile[c * LDS_STRIDE + q * 4];
        dst[0] = v[0]; dst[1] = v[1]; dst[2] = v[2]; dst[3] = v[3];
      }
    }
  }
  __syncthreads();

  // ------------- transform + store phase: coalesced NT writes along c --------
  // out flat index: ((b*G2 + g)*NCHAN + c), c = a*85 + attr.
  const size_t obase = (size_t)b * G2 * NCHAN;
  for (int idx = t; idx < TBG * NCHAN; idx += BLOCK) {
    const int gl = idx / NCHAN;         // magic-number division (const divisor)
    const int c  = idx - gl * NCHAN;
    const int g  = g_base + gl;
    if (g >= G2) break;                 // gl monotonically nondecreasing in idx

    const float v    = tile[c * LDS_STRIDE + gl];
    const int   a    = c / NATTR;
    const int   attr = c - a * NATTR;

    float r;
    if (attr >= 4) {
      // objectness + class scores: sigmoid(v) = rcp(1 + exp2(-log2e*v))
      r = __builtin_amdgcn_rcpf(1.0f + __builtin_amdgcn_exp2f(-LOG2E * v));
    } else if (attr < 2) {
      const int   gy  = g / GDIM;       // meshgrid: gx = col, gy = row
      const int   gx  = g - gy * GDIM;
      const float off = (attr == 0) ? (float)gx : (float)gy;
      const float s =
          __builtin_amdgcn_rcpf(1.0f + __builtin_amdgcn_exp2f(-LOG2E * v));
      r = (s + off) * STRIDE_F;
    } else {
      // wh: exp(v) * (ANCHORS[a]/stride) * stride = exp(v) * ANCHORS[a][dim]
      const float anch =
          (attr == 2) ? ((a == 0) ? 10.0f : (a == 1) ? 16.0f : 33.0f)
                      : ((a == 0) ? 13.0f : (a == 1) ? 30.0f : 23.0f);
      r = __builtin_amdgcn_exp2f(LOG2E * v) * anch;
    }
    __builtin_nontemporal_store(r, out + obase + (size_t)g * NCHAN + c);
  }
}

extern "C" void kernel_launch(void* const* d_in, const int* in_sizes, int n_in,
                              void* d_out, int out_size, void* d_ws, size_t ws_size,
                              hipStream_t stream) {
  const float* x = (const float*)d_in[0];
  float* out     = (float*)d_out;
  const int B    = in_sizes[0] / (NCHAN * G2);  // 16 for the reference shapes

  dim3 grid((G2 + TBG - 1) / TBG, B);           // 181 x B blocks
  yolo_head_kernel<<<grid, BLOCK, 0, stream>>>(x, out, B);
}